// DeeperGCN_76794015252782
// MI455X (gfx1250) — compile-verified
//
#include <hip/hip_runtime.h>
#include <hip/hip_bf16.h>

typedef __attribute__((ext_vector_type(2))) float v2f;
typedef __attribute__((ext_vector_type(8))) float v8f;

#define HCH 64
#define EPS_GEN 1e-7f
#define EPS_BN  1e-5f

// ---------------------------------------------------------------------------
// Generic fills / copies
// ---------------------------------------------------------------------------
__global__ __launch_bounds__(256) void fill_f32(float* p, float v, int n) {
    int i = blockIdx.x * 256 + threadIdx.x;
    if (i < n) p[i] = v;
}

__global__ __launch_bounds__(256) void copy_f32(float* dst, const float* src, int n) {
    int i = blockIdx.x * 256 + threadIdx.x;
    if (i < n) dst[i] = src[i];
}

__global__ __launch_bounds__(256) void vn_bcast(const float* vn0, float* vn, int n) {
    int i = blockIdx.x * 256 + threadIdx.x;
    if (i < n) vn[i] = vn0[i & (HCH - 1)];
}

// ---------------------------------------------------------------------------
// Atom encoder: h[i,c] = sum_f atom_table[ATOM_OFF[f]+x[i,f], c] + vn0[c]
// ---------------------------------------------------------------------------
__global__ __launch_bounds__(256) void atom_encode(
    const float* __restrict__ atom_table, const int* __restrict__ x,
    const float* __restrict__ vn0, float* __restrict__ h, int nrows)
{
    const int AOFF[9] = {0, 119, 124, 136, 148, 158, 164, 170, 172};
    int idx = blockIdx.x * 256 + threadIdx.x;
    int row = idx >> 6, c = idx & (HCH - 1);
    if (row >= nrows) return;
    float s = vn0[c];
    #pragma unroll
    for (int f = 0; f < 9; ++f) {
        int r = AOFF[f] + x[row * 9 + f];
        s += atom_table[r * HCH + c];
    }
    h[idx] = s;
}

// ---------------------------------------------------------------------------
// GENConv message + scatter-add: m[dst] += relu(h[src] + bond_emb) + eps
// bond table (13x64 floats) is cache-resident; recompute emb per edge.
// ---------------------------------------------------------------------------
__global__ __launch_bounds__(256) void msg_scatter(
    const float* __restrict__ h, const float* __restrict__ bond_table,
    const int* __restrict__ src, const int* __restrict__ dst,
    const int* __restrict__ eattr, float* __restrict__ m, int nedges)
{
    int idx = blockIdx.x * 256 + threadIdx.x;
    int e = idx >> 6, c = idx & (HCH - 1);
    if (e >= nedges) return;
    int s = src[e], d = dst[e];
    int a0 = eattr[e * 3 + 0], a1 = eattr[e * 3 + 1], a2 = eattr[e * 3 + 2];
    float emb = bond_table[(0 + a0) * HCH + c]
              + bond_table[(5 + a1) * HCH + c]
              + bond_table[(11 + a2) * HCH + c];
    float v = h[(size_t)s * HCH + c] + emb;
    v = fmaxf(v, 0.0f) + EPS_GEN;
    atomicAdd(&m[(size_t)d * HCH + c], v);
}

// ---------------------------------------------------------------------------
// WMMA f32 GEMM: out[rows,64] = (A1 (+A2)) @ W[64,64] + bias (+ res)
// One 256-thread block = 8 waves; each wave -> 16 rows x 64 cols,
// 4 f32 accumulators, K fully unrolled with V_WMMA_F32_16X16X4_F32.
// A2/res fusion are template params (no runtime branch in the hot loop).
// W is staged in LDS K-pair interleaved: Wp[k>>1][col*2 + (k&1)], so each
// lane's B fragment (W[kA,col], W[kA+1,col]) is one aligned ds_load_b64.
// ---------------------------------------------------------------------------
template <bool HAS_A2, bool HAS_RES>
__global__ __launch_bounds__(256) void gemm64_wmma(
    const float* __restrict__ A1, const float* __restrict__ A2,
    const float* __restrict__ W, const float* __restrict__ bias,
    const float* __restrict__ res, float* __restrict__ out, int rows)
{
    __shared__ float Wp[(HCH / 2) * HCH * 2];   // [32][64][2] = 4096 floats
    __shared__ float bl[HCH];
    for (int i = threadIdx.x; i < HCH * HCH; i += 256) {
        int k = i >> 6, c = i & (HCH - 1);
        Wp[(k >> 1) * (HCH * 2) + c * 2 + (k & 1)] = W[i];
    }
    if (threadIdx.x < HCH) bl[threadIdx.x] = bias[threadIdx.x];
    __syncthreads();

    const int lane = threadIdx.x & 31;
    const int wave = threadIdx.x >> 5;
    const int row0 = blockIdx.x * 128 + wave * 16;
    const int nl   = lane & 15;
    const int half = lane >> 4;       // 0 or 1
    const int koff = half * 2;        // A/B 16x4 f32 fragment layout (kA even)

    int arow = row0 + nl;
    if (arow >= rows) arow = rows - 1;          // clamp; stores are guarded
    const float* a1p = A1 + (size_t)arow * HCH;
    const float* a2p = HAS_A2 ? (A2 + (size_t)arow * HCH) : nullptr;

    v8f acc[4] = {};

    #pragma unroll
    for (int k0 = 0; k0 < HCH; k0 += 4) {
        const int kA = k0 + koff;               // always even
        v2f a = *(const v2f*)&a1p[kA];
        if (HAS_A2) a = a + *(const v2f*)&a2p[kA];
        const float* bp = &Wp[(kA >> 1) * (HCH * 2) + nl * 2];
        #pragma unroll
        for (int t = 0; t < 4; ++t) {
            v2f b = *(const v2f*)&bp[t * 32];   // col = t*16 + nl
            acc[t] = __builtin_amdgcn_wmma_f32_16x16x4_f32(
                false, a, false, b, (short)0, acc[t], false, false);
        }
    }

    #pragma unroll
    for (int r = 0; r < 8; ++r) {
        int rr = row0 + r + half * 8;           // C layout: vgpr r -> M=r / M=r+8
        if (rr < rows) {
            #pragma unroll
            for (int t = 0; t < 4; ++t) {
                int col = t * 16 + nl;
                float v = acc[t][r] + bl[col];
                if (HAS_RES) v += res[(size_t)rr * HCH + col];
                out[(size_t)rr * HCH + col] = v;
            }
        }
    }
}

// ---------------------------------------------------------------------------
// BatchNorm statistics: stats[0:64]=sum, stats[64:128]=sum of squares
// ---------------------------------------------------------------------------
__global__ __launch_bounds__(256) void bn_stats(
    const float* __restrict__ x, float* __restrict__ stats, int rows)
{
    __shared__ float ls[HCH], lq[HCH];
    int c = threadIdx.x & (HCH - 1);
    int sub = threadIdx.x >> 6;   // 0..3
    if (threadIdx.x < HCH) { ls[threadIdx.x] = 0.0f; lq[threadIdx.x] = 0.0f; }
    __syncthreads();
    float s = 0.0f, q = 0.0f;
    for (int r = blockIdx.x * 4 + sub; r < rows; r += gridDim.x * 4) {
        float v = x[(size_t)r * HCH + c];
        s += v; q += v * v;
    }
    atomicAdd(&ls[c], s);
    atomicAdd(&lq[c], q);
    __syncthreads();
    if (threadIdx.x < HCH) {
        atomicAdd(&stats[threadIdx.x], ls[threadIdx.x]);
        atomicAdd(&stats[HCH + threadIdx.x], lq[threadIdx.x]);
    }
}

// ---------------------------------------------------------------------------
// BN apply (+optional ReLU, +optional fused segment-sum into seg[batch[row]])
// ---------------------------------------------------------------------------
__global__ __launch_bounds__(256) void bn_apply(
    const float* __restrict__ x, const float* __restrict__ stats,
    const float* __restrict__ g, const float* __restrict__ be,
    float* __restrict__ out, float* __restrict__ seg,
    const int* __restrict__ batch, int rows, int do_relu)
{
    int idx = blockIdx.x * 256 + threadIdx.x;
    int row = idx >> 6, c = idx & (HCH - 1);
    if (row >= rows) return;
    float inv = 1.0f / (float)rows;
    float mu  = stats[c] * inv;
    float var = stats[HCH + c] * inv - mu * mu;
    float rs  = rsqrtf(var + EPS_BN);
    float v = (x[idx] - mu) * rs * g[c] + be[c];
    if (do_relu) v = fmaxf(v, 0.0f);
    out[idx] = v;
    if (seg) atomicAdd(&seg[(size_t)batch[row] * HCH + c], v);
}

// ---------------------------------------------------------------------------
// h2[i] += vn[batch[i]]
// ---------------------------------------------------------------------------
__global__ __launch_bounds__(256) void add_vn(
    float* __restrict__ h2, const float* __restrict__ vn,
    const int* __restrict__ batch, int rows)
{
    int idx = blockIdx.x * 256 + threadIdx.x;
    int row = idx >> 6, c = idx & (HCH - 1);
    if (row >= rows) return;
    h2[idx] += vn[(size_t)batch[row] * HCH + c];
}

// ---------------------------------------------------------------------------
// Mean-pool readout
// ---------------------------------------------------------------------------
__global__ __launch_bounds__(256) void pool_sum(
    const float* __restrict__ h, const int* __restrict__ batch,
    float* __restrict__ sums, float* __restrict__ cnt, int rows)
{
    int idx = blockIdx.x * 256 + threadIdx.x;
    int row = idx >> 6, c = idx & (HCH - 1);
    if (row >= rows) return;
    int g = batch[row];
    atomicAdd(&sums[(size_t)g * HCH + c], h[idx]);
    if (c == 0) atomicAdd(&cnt[g], 1.0f);
}

__global__ __launch_bounds__(256) void pool_div(
    const float* __restrict__ sums, const float* __restrict__ cnt,
    float* __restrict__ out, int n)
{
    int idx = blockIdx.x * 256 + threadIdx.x;
    if (idx >= n) return;
    out[idx] = sums[idx] / fmaxf(cnt[idx >> 6], 1.0f);
}

// ---------------------------------------------------------------------------
// Host orchestration
// ---------------------------------------------------------------------------
static inline int cdiv(int a, int b) { return (a + b - 1) / b; }

extern "C" void kernel_launch(void* const* d_in, const int* in_sizes, int n_in,
                              void* d_out, int out_size, void* d_ws, size_t ws_size,
                              hipStream_t stream) {
    const float* atom_table = (const float*)d_in[0];
    const float* bond_table = (const float*)d_in[1];
    const float* vn0        = (const float*)d_in[2];
    const float* gcn_W      = (const float*)d_in[3];   // [7,64,64]
    const float* gcn_b      = (const float*)d_in[4];   // [7,64]
    const float* gamma      = (const float*)d_in[5];
    const float* beta       = (const float*)d_in[6];
    const float* vn_W1      = (const float*)d_in[7];   // [6,64,64]
    const float* vn_b1      = (const float*)d_in[8];
    const float* vn_g1      = (const float*)d_in[9];
    const float* vn_be1     = (const float*)d_in[10];
    const float* vn_W2      = (const float*)d_in[11];
    const float* vn_b2      = (const float*)d_in[12];
    const float* vn_g2      = (const float*)d_in[13];
    const float* vn_be2     = (const float*)d_in[14];
    const int*   x          = (const int*)d_in[15];    // [N,9]
    const int*   edge_index = (const int*)d_in[16];    // [2,E]
    const int*   eattr      = (const int*)d_in[17];    // [E,3]
    const int*   batch      = (const int*)d_in[18];    // [N]

    const int N = in_sizes[15] / 9;
    const int E = in_sizes[16] / 2;
    const int G = out_size / HCH;
    const int L = 7;
    const int N64 = N * HCH;
    const int G64 = G * HCH;

    const int* src = edge_index;
    const int* dst = edge_index + E;

    float* ws   = (float*)d_ws;
    float* h    = ws;                 // [N,64]
    float* m    = h  + N64;           // [N,64]
    float* h2   = m  + N64;           // [N,64]
    float* vn   = h2 + N64;           // [G,64]
    float* vt   = vn + G64;           // [G,64]
    float* z    = vt + G64;           // [G,64]
    float* zt   = z  + G64;           // [G,64]
    float* stat = zt + G64;           // [128]
    float* gcnt = stat + 2 * HCH;     // [G]
    (void)ws_size; (void)n_in;

    const int gN64 = cdiv(N64, 256);
    const int gE64 = cdiv(E * HCH, 256);
    const int gG64 = cdiv(G64, 256);
    const int gGemmN = cdiv(N, 128);
    const int gGemmG = cdiv(G, 128);
    const int gStatN = 256;
    const int gStatG = cdiv(G, 4);

    // ---- encoders + initial state ------------------------------------
    atom_encode<<<gN64, 256, 0, stream>>>(atom_table, x, vn0, h, N);
    vn_bcast<<<gG64, 256, 0, stream>>>(vn0, vn, G64);

    // ---- layer 0: h = (h + scatter(msg)) @ W0 + b0 -------------------
    fill_f32<<<gN64, 256, 0, stream>>>(m, 0.0f, N64);
    msg_scatter<<<gE64, 256, 0, stream>>>(h, bond_table, src, dst, eattr, m, E);
    gemm64_wmma<true, false><<<gGemmN, 256, 0, stream>>>(
        h, m, gcn_W, gcn_b, nullptr, h, N);

    // ---- layers 1..6 (res+ blocks with virtual node) -----------------
    for (int l = 1; l < L; ++l) {
        const int j = l - 1;
        // h2 = relu(BN(h)); vt = vn + segment_sum(h2, batch)
        fill_f32<<<1, 256, 0, stream>>>(stat, 0.0f, 2 * HCH);
        bn_stats<<<gStatN, 256, 0, stream>>>(h, stat, N);
        copy_f32<<<gG64, 256, 0, stream>>>(vt, vn, G64);
        bn_apply<<<gN64, 256, 0, stream>>>(h, stat, gamma + j * HCH, beta + j * HCH,
                                           h2, vt, batch, N, 1);
        // z = relu(BN(vt @ W1 + b1))
        gemm64_wmma<false, false><<<gGemmG, 256, 0, stream>>>(
            vt, nullptr, vn_W1 + j * HCH * HCH, vn_b1 + j * HCH, nullptr, zt, G);
        fill_f32<<<1, 256, 0, stream>>>(stat, 0.0f, 2 * HCH);
        bn_stats<<<gStatG, 256, 0, stream>>>(zt, stat, G);
        bn_apply<<<gG64, 256, 0, stream>>>(zt, stat, vn_g1 + j * HCH, vn_be1 + j * HCH,
                                           z, nullptr, nullptr, G, 1);
        // vn = relu(BN(z @ W2 + b2))
        gemm64_wmma<false, false><<<gGemmG, 256, 0, stream>>>(
            z, nullptr, vn_W2 + j * HCH * HCH, vn_b2 + j * HCH, nullptr, zt, G);
        fill_f32<<<1, 256, 0, stream>>>(stat, 0.0f, 2 * HCH);
        bn_stats<<<gStatG, 256, 0, stream>>>(zt, stat, G);
        bn_apply<<<gG64, 256, 0, stream>>>(zt, stat, vn_g2 + j * HCH, vn_be2 + j * HCH,
                                           vn, nullptr, nullptr, G, 1);
        // h2 += vn[batch]; h = (h2 + scatter(msg(h2))) @ Wl + bl + h
        add_vn<<<gN64, 256, 0, stream>>>(h2, vn, batch, N);
        fill_f32<<<gN64, 256, 0, stream>>>(m, 0.0f, N64);
        msg_scatter<<<gE64, 256, 0, stream>>>(h2, bond_table, src, dst, eattr, m, E);
        gemm64_wmma<true, true><<<gGemmN, 256, 0, stream>>>(
            h2, m, gcn_W + l * HCH * HCH, gcn_b + l * HCH, h, h, N);
    }

    // ---- final BN + global mean pool ---------------------------------
    fill_f32<<<1, 256, 0, stream>>>(stat, 0.0f, 2 * HCH);
    bn_stats<<<gStatN, 256, 0, stream>>>(h, stat, N);
    bn_apply<<<gN64, 256, 0, stream>>>(h, stat, gamma + (L - 1) * HCH,
                                       beta + (L - 1) * HCH, h2, nullptr, nullptr, N, 0);
    fill_f32<<<gG64, 256, 0, stream>>>(vt, 0.0f, G64);
    fill_f32<<<cdiv(G, 256), 256, 0, stream>>>(gcnt, 0.0f, G);
    pool_sum<<<gN64, 256, 0, stream>>>(h2, batch, vt, gcnt, N);
    pool_div<<<gG64, 256, 0, stream>>>(vt, gcnt, (float*)d_out, G64);
}